// DynamicPatch_66030827209030
// MI455X (gfx1250) — compile-verified
//
#include <hip/hip_runtime.h>

#define PATCH  32
#define LEVELS 3
#define NEXP   4
#define NREAL  3

typedef __attribute__((ext_vector_type(2))) float v2f;
typedef __attribute__((ext_vector_type(8))) float v8f;

// ---------------------------------------------------------------------------
// Kernel 1: gate GEMM (v_wmma_f32_16x16x4_f32) + softmax + stable top-2.
// Computes D = Wg (4x32, zero-padded to 16 rows) x x_tile^T (32 x 16 patches),
// accumulated over K=32 in 8 WMMA steps of K=4.  Lane m (0..15) of each wave
// ends with all 4 expert logits of patch tile*16+m in C components 0..3.
// A-operand zero-padding is done branchlessly: clamped row pointer +
// multiply-by-mask, so EXEC stays all-ones and loads are plain b64s.
// ---------------------------------------------------------------------------
__global__ void gate_kernel(const float* __restrict__ x,
                            const float* __restrict__ Wg,
                            int*    __restrict__ lvlA,
                            int2*   __restrict__ idxA,
                            float2* __restrict__ wA,
                            int tiles) {
  const int gtid = blockIdx.x * blockDim.x + threadIdx.x;
  const int wave = gtid >> 5;                 // one wave32 per 16-patch tile
  const int lane = threadIdx.x & 31;
  if (wave >= tiles) return;                  // wave-uniform guard

  const int m  = lane & 15;                   // A row / B column index
  const int kb = (lane >> 4) << 1;            // K sub-offset: 0 (lanes<16) or 2

  const float amask = (m < NEXP) ? 1.0f : 0.0f;          // zero rows 4..15
  const int   mrow  = (m < NEXP) ? m : 0;                // clamped (in-bounds)
  const float* __restrict__ xrow = x  + (size_t)(wave * 16 + m) * PATCH;
  const float* __restrict__ wrow = Wg + (size_t)mrow * PATCH;

  v8f c = {0.f, 0.f, 0.f, 0.f, 0.f, 0.f, 0.f, 0.f};
#pragma unroll
  for (int s = 0; s < 8; ++s) {
    const int k = 4 * s + kb;                 // even -> 8B-aligned float2 loads
    const float2 av = *(const float2*)(wrow + k);
    const float2 bv = *(const float2*)(xrow + k);
    v2f a, b;
    a.x = av.x * amask;                       // branchless zero-padding
    a.y = av.y * amask;
    b.x = bv.x;
    b.y = bv.y;
    c = __builtin_amdgcn_wmma_f32_16x16x4_f32(false, a, false, b,
                                              (short)0, c, false, false);
  }

  if (lane < 16) {
    const float l0 = c[0], l1 = c[1], l2 = c[2], l3 = c[3];
    const float mx = fmaxf(fmaxf(l0, l1), fmaxf(l2, l3));
    const float e0 = __expf(l0 - mx), e1 = __expf(l1 - mx),
                e2 = __expf(l2 - mx), e3 = __expf(l3 - mx);
    const float inv = 1.0f / (e0 + e1 + e2 + e3);
    float sc[4] = {e0 * inv, e1 * inv, e2 * inv, e3 * inv};

    // stable descending top-2 (ties -> lower expert index, matches argsort)
    int b0 = 0;
#pragma unroll
    for (int e = 1; e < 4; ++e) if (sc[e] > sc[b0]) b0 = e;
    int b1 = (b0 == 0) ? 1 : 0;
#pragma unroll
    for (int e = 0; e < 4; ++e) if (e != b0 && sc[e] > sc[b1]) b1 = e;

    const int m0 = (b0 < NREAL) ? b0 : -1;
    const int m1 = (b1 < NREAL) ? b1 : -1;
    int lv = m0 > m1 ? m0 : m1;
    if (lv < 0) lv = 0;

    const int p = wave * 16 + lane;
    lvlA[p] = lv;
    idxA[p] = make_int2(b0, b1);
    wA[p]   = make_float2(sc[b0], sc[b1]);    // ROUTE_SCALE == 1
  }
}

// ---------------------------------------------------------------------------
// Kernel 2: exclusive prefix sum of counts (2^lvl) over all P patches.
// Single block of 1024 threads: serial chunk sums + Hillis-Steele in LDS.
// ---------------------------------------------------------------------------
__global__ void scan_kernel(const int* __restrict__ lvlA,
                            int* __restrict__ startsA, int P) {
  __shared__ int part[1024];
  const int tid  = threadIdx.x;
  const int ch   = P >> 10;                   // patches per thread (P % 1024 == 0)
  const int base = tid * ch;

  int s = 0;
  for (int i = 0; i < ch; ++i) s += 1 << lvlA[base + i];
  part[tid] = s;
  __syncthreads();

  for (int off = 1; off < 1024; off <<= 1) {
    int v = (tid >= off) ? part[tid - off] : 0;
    __syncthreads();
    part[tid] += v;
    __syncthreads();
  }

  int run = (tid == 0) ? 0 : part[tid - 1];   // exclusive prefix for this chunk
  for (int i = 0; i < ch; ++i) {
    startsA[base + i] = run;
    run += 1 << lvlA[base + i];
  }
}

// ---------------------------------------------------------------------------
// Kernel 3: one wave32 per parent patch; scatter children into flat outputs.
// Output layout (floats): new_x[32T] | new_mask[32T] | new_size[T] |
//                         weights[2T] | expert_idx[2T] | x_final[96T]
// ---------------------------------------------------------------------------
__global__ void emit_kernel(const float* __restrict__ x,
                            const float* __restrict__ mask,
                            const int*    __restrict__ lvlA,
                            const int*    __restrict__ startsA,
                            const int2*   __restrict__ idxA,
                            const float2* __restrict__ wA,
                            float* __restrict__ out,
                            long long T, int P) {
  const int warp = (blockIdx.x * blockDim.x + threadIdx.x) >> 5;
  const int lane = threadIdx.x & 31;
  if (warp >= P) return;

  const int    p     = warp;
  const int    lv    = lvlA[p];
  const int    start = startsA[p];
  const int2   id    = idxA[p];
  const float2 w     = wA[p];
  const float  xv    = x[(size_t)p * PATCH + lane];
  const float  mv    = mask[(size_t)p * PATCH + lane];
  const int    nch   = 1 << lv;
  const int    sz    = PATCH >> lv;
  const float  xm    = xv * mv;

  float* outx = out;
  float* outm = out + 32LL * T;
  float* outs = out + 64LL * T;
  float* outw = out + 65LL * T;
  float* oute = out + 67LL * T;
  float* outf = out + 69LL * T;

  for (int cI = 0; cI < nch; ++cI) {
    const long long t   = start + cI;
    const int       src = (cI * sz + lane) & 31;         // gather pos in parent
    const float gx = __shfl(xv, src, 32);
    const float gm = __shfl(mv, src, 32);
    const bool valid = lane < sz;

    outx[t * 32 + lane] = valid ? gx : 0.f;
    outm[t * 32 + lane] = valid ? gm : 0.f;

    if (lane == 0) {
      outs[t]         = (float)sz;
      outw[2 * t]     = w.x;
      outw[2 * t + 1] = w.y;
      oute[2 * t]     = (float)id.x;
      oute[2 * t + 1] = (float)id.y;
    }

#pragma unroll
    for (int l = 0; l < LEVELS; ++l) {
      float val = 0.f;
      if (l == lv) {                                     // own chunk at own level
        if (lane >= cI * sz && lane < (cI + 1) * sz) val = xm;
      } else if (lv >= 1 && l == lv - 1) {               // containing coarser chunk
        const int cp  = cI >> 1;
        const int szp = PATCH >> (lv - 1);
        if (lane >= cp * szp && lane < (cp + 1) * szp) val = xm;
      }
      outf[(t * 3 + l) * 32 + lane] = val;
    }
  }
}

// ---------------------------------------------------------------------------
extern "C" void kernel_launch(void* const* d_in, const int* in_sizes, int n_in,
                              void* d_out, int out_size, void* d_ws, size_t ws_size,
                              hipStream_t stream) {
  const float* x    = (const float*)d_in[0];
  const float* mask = (const float*)d_in[1];
  const float* Wg   = (const float*)d_in[2];

  const int       P = in_sizes[0] / PATCH;          // 65536
  const long long T = (long long)out_size / 165;    // total children

  char*   ws      = (char*)d_ws;
  int*    lvlA    = (int*)   (ws);
  int*    startsA = (int*)   (ws + (size_t)P * 4);
  int2*   idxA    = (int2*)  (ws + (size_t)P * 8);
  float2* wA      = (float2*)(ws + (size_t)P * 16);

  const int tiles      = P / 16;                    // 16 patches per wave32
  const int gateBlocks = (tiles * 32 + 255) / 256;
  gate_kernel<<<gateBlocks, 256, 0, stream>>>(x, Wg, lvlA, idxA, wA, tiles);

  scan_kernel<<<1, 1024, 0, stream>>>(lvlA, startsA, P);

  const int emitBlocks = (P * 32 + 255) / 256;      // one wave per parent
  emit_kernel<<<emitBlocks, 256, 0, stream>>>(x, mask, lvlA, startsA, idxA, wA,
                                              (float*)d_out, T, P);
}